// GCN_80977313399075
// MI455X (gfx1250) — compile-verified
//
#include <hip/hip_runtime.h>
#include <hip/hip_bf16.h>

#define N_NODES  50000
#define N_EDGES  1600000
#define NFEAT    256
#define HIDDEN   256
#define NCLASS   32
#define N_GRAPHS 64

typedef __attribute__((ext_vector_type(2))) float v2f;
typedef __attribute__((ext_vector_type(8))) float v8f;
typedef __attribute__((ext_vector_type(4))) unsigned int u32x4;
typedef __attribute__((ext_vector_type(8))) unsigned int u32x8;

// Padded LDS row stride (floats): 256 + 2 so that the 16 lanes of a WMMA
// A-fragment ds_load_b64 hit 16 distinct banks (258 mod 64 == 2).
// The +2 padding is produced by the TDM itself (pad_interval=256 dwords,
// pad_amount=2 dwords), so the DMA lays rows out pre-swizzled.
#define APAD 258

// ---------------------------------------------------------------------------
// GEMM C[M,N] = op(A[M,K]) @ B[K,N], fp32.
//  * One workgroup owns a 16-row M-slab. Wave 0 issues a single
//    tensor_load_to_lds (TDM DMA) that pulls the whole 16xK fp32 slab into
//    LDS (with bank-conflict padding), waits on TENSORcnt, then the block
//    barriers. A is read once from global per slab instead of once per
//    N-tile (16x less L2 traffic for layer 1).
//  * Each wave computes NT 16x16 output tiles with V_WMMA_F32_16X16X4_F32,
//    A-fragments via ds_load_b64 from LDS, B-fragments from global
//    (W1/W2 are 256KB/32KB -> L2/L0 resident).
//  * Grid/block sized so tiles divide exactly -> EXEC all-1s at every WMMA.
//
// f32 16x16x4 fragment layouts (ISA 7.12.2):
//   A: lane<16 -> {K0,K1}, lane>=16 -> {K2,K3}; M = lane&15
//   B: mirrored with N = lane&15
//   C: VGPR p: lanes0-15 -> M=p, lanes16-31 -> M=p+8; N = lane&15
// ---------------------------------------------------------------------------
template<int RELU, int NT>
__global__ __launch_bounds__(256) void gemm_wmma_tdm(
    const float* __restrict__ A, const float* __restrict__ B,
    float* __restrict__ C, int N, int K)
{
  __shared__ float sA[16][APAD];

  const int lane = threadIdx.x & 31;
  const int wave = threadIdx.x >> 5;
  const int half = lane >> 4;
  const int l    = lane & 15;
  const int row0 = blockIdx.x * 16;

  if (wave == 0) {
    // Build the Tensor DMA descriptor (D#), groups 0 and 1 (2D tensor,
    // VADDR2/VADDR3 disabled). All fields per CDNA5 ISA 8.3/8.4.
    unsigned long long ga = (unsigned long long)(const void*)(A + (size_t)row0 * K);
    unsigned lds = (unsigned)(size_t)(void*)&sA[0][0];

    u32x4 g0;
    g0[0] = 1u;                                            // count=1, user mode
    g0[1] = lds;                                           // lds_addr
    g0[2] = (unsigned)ga;                                  // global_addr[31:0]
    g0[3] = (unsigned)((ga >> 32) & 0x01FFFFFFull)         // global_addr[56:32]
          | (2u << 30);                                    // type = 2 ("image")

    u32x8 g1;
    g1[0] = (2u << 16)                                     // data_size = 4 bytes
          | (1u << 20)                                     // pad_enable
          | (7u << 22)                                     // pad_interval = 256 dw
          | (1u << 25);                                    // pad_amount   = 2 dw
    g1[1] = ((unsigned)K & 0xFFFFu) << 16;                 // tensor_dim0 lo16
    g1[2] = ((unsigned)K >> 16) | (16u << 16);             // dim0 hi | tensor_dim1=16
    g1[3] = ((unsigned)K & 0xFFFFu) << 16;                 // dim1 hi=0 | tile_dim0=K
    g1[4] = 16u;                                           // tile_dim1=16, tile_dim2=0
    g1[5] = (unsigned)K;                                   // tensor_dim0_stride lo
    g1[6] = 0u;                                            // stride hi | dim1_stride lo
    g1[7] = 0u;                                            // dim1_stride hi

    asm volatile("tensor_load_to_lds %0, %1" :: "s"(g0), "s"(g1) : "memory");
    __builtin_amdgcn_s_wait_tensorcnt(0);
  }
  __syncthreads();

  const int koff = half * 2;
  const int tn0  = wave * NT;

  v8f acc[NT];
#pragma unroll
  for (int t = 0; t < NT; ++t) acc[t] = (v8f){};

  for (int k = 0; k < K; k += 4) {
    v2f a = *(const v2f*)&sA[l][k + koff];                 // ds_load_b64
    if (RELU) { a.x = fmaxf(a.x, 0.0f); a.y = fmaxf(a.y, 0.0f); }
#pragma unroll
    for (int t = 0; t < NT; ++t) {
      const int col = (tn0 + t) * 16 + l;
      v2f b;
      b.x = B[(size_t)(k + koff)     * N + col];
      b.y = B[(size_t)(k + koff + 1) * N + col];
      acc[t] = __builtin_amdgcn_wmma_f32_16x16x4_f32(
          false, a, false, b, (short)0, acc[t], false, false);
    }
  }

  float* Crow0 = C + (size_t)(row0 + half * 8) * N;
#pragma unroll
  for (int t = 0; t < NT; ++t) {
    const int col = (tn0 + t) * 16 + l;
#pragma unroll
    for (int p = 0; p < 8; ++p) Crow0[(size_t)p * N + col] = acc[t][p];
  }
}

// ---------------------------------------------------------------------------
// Edge scatter-add: out[dst[e], :] += h[src[e], :]. One thread per
// (edge, float4 chunk): b128 gather (stays in 192MB L2: h fits) and
// 4x global_atomic_add_f32 handled by the L2 atomic units.
// edge_index is [2, E] flat: src = ei[e], dst = ei[E + e].
// ---------------------------------------------------------------------------
template<int F>
__global__ __launch_bounds__(256) void scatter_add4(
    const float* __restrict__ h, const int* __restrict__ ei,
    float* __restrict__ out, int E)
{
  constexpr int F4 = F >> 2;
  long long tid = (long long)blockIdx.x * blockDim.x + threadIdx.x;
  if (tid >= (long long)E * F4) return;
  int e = (int)(tid / F4);
  int c = (int)(tid - (long long)e * F4);
  int s = ei[e];
  int d = ei[E + e];
  const float4 v = ((const float4*)(h + (size_t)s * F))[c];
  float* o = out + (size_t)d * F + (size_t)c * 4;
  atomicAdd(o + 0, v.x);
  atomicAdd(o + 1, v.y);
  atomicAdd(o + 2, v.z);
  atomicAdd(o + 3, v.w);
}

// ---------------------------------------------------------------------------
// Mean pool: accumulate per-graph sums + node counts, then divide.
// ---------------------------------------------------------------------------
__global__ __launch_bounds__(256) void pool_accum(
    const float* __restrict__ h, const int* __restrict__ batch,
    float* __restrict__ sums, float* __restrict__ counts)
{
  int tid = blockIdx.x * blockDim.x + threadIdx.x;
  if (tid >= N_NODES * NCLASS) return;
  int node = tid >> 5;                 // NCLASS == 32
  int f    = tid & 31;
  int g    = batch[node];
  atomicAdd(&sums[g * NCLASS + f], h[tid]);
  if (f == 0) atomicAdd(&counts[g], 1.0f);
}

__global__ void pool_final(const float* __restrict__ sums,
                           const float* __restrict__ counts,
                           float* __restrict__ out)
{
  int tid = blockIdx.x * blockDim.x + threadIdx.x;
  if (tid >= N_GRAPHS * NCLASS) return;
  int g = tid >> 5;
  out[tid] = sums[tid] / fmaxf(counts[g], 1.0f);
}

// ---------------------------------------------------------------------------
extern "C" void kernel_launch(void* const* d_in, const int* in_sizes, int n_in,
                              void* d_out, int out_size, void* d_ws, size_t ws_size,
                              hipStream_t stream)
{
  const float* x     = (const float*)d_in[0];   // [N_NODES, NFEAT]
  const int*   ei    = (const int*)  d_in[1];   // [2, N_EDGES]
  const int*   batch = (const int*)  d_in[2];   // [N_NODES]
  const float* W1    = (const float*)d_in[3];   // [NFEAT, HIDDEN]
  const float* W2    = (const float*)d_in[4];   // [HIDDEN, NCLASS]

  char* ws = (char*)d_ws;
  float* h1     = (float*)ws; ws += (size_t)N_NODES * HIDDEN * sizeof(float);
  float* agg1   = (float*)ws; ws += (size_t)N_NODES * HIDDEN * sizeof(float);
  float* h2     = (float*)ws; ws += (size_t)N_NODES * NCLASS * sizeof(float);
  float* agg2   = (float*)ws; ws += (size_t)N_NODES * NCLASS * sizeof(float);
  float* sums   = (float*)ws; ws += (size_t)N_GRAPHS * NCLASS * sizeof(float);
  float* counts = (float*)ws;

  hipMemsetAsync(agg1,   0, (size_t)N_NODES * HIDDEN * sizeof(float), stream);
  hipMemsetAsync(agg2,   0, (size_t)N_NODES * NCLASS * sizeof(float), stream);
  hipMemsetAsync(sums,   0, (size_t)N_GRAPHS * NCLASS * sizeof(float), stream);
  hipMemsetAsync(counts, 0, (size_t)N_GRAPHS * sizeof(float), stream);

  // Layer 1 GEMM: h1 = x @ W1. 3125 blocks x 256 threads = 8 waves,
  // 2 N-tiles per wave -> 16 N-tiles = full N=256 (exact coverage).
  gemm_wmma_tdm<0, 2><<<3125, 256, 0, stream>>>(x, W1, h1, HIDDEN, NFEAT);

  // Aggregate: agg1[dst] += h1[src]
  {
    long long total = (long long)N_EDGES * (HIDDEN / 4);
    int blocks = (int)((total + 255) / 256);
    scatter_add4<HIDDEN><<<blocks, 256, 0, stream>>>(h1, ei, agg1, N_EDGES);
  }

  // Layer 2 GEMM: h2 = relu(agg1) @ W2. 3125 blocks x 64 threads = 2 waves,
  // 1 N-tile per wave -> 2 N-tiles = full N=32 (exact coverage).
  gemm_wmma_tdm<1, 1><<<3125, 64, 0, stream>>>(agg1, W2, h2, NCLASS, HIDDEN);

  // Aggregate: agg2[dst] += h2[src]
  {
    long long total = (long long)N_EDGES * (NCLASS / 4);
    int blocks = (int)((total + 255) / 256);
    scatter_add4<NCLASS><<<blocks, 256, 0, stream>>>(h2, ei, agg2, N_EDGES);
  }

  // Global mean pool -> d_out [64, 32]
  pool_accum<<<(N_NODES * NCLASS + 255) / 256, 256, 0, stream>>>(agg2, batch, sums, counts);
  pool_final<<<(N_GRAPHS * NCLASS + 255) / 256, 256, 0, stream>>>(sums, counts, (float*)d_out);
}